// MLMM_electrostatics_NoShift_48498770706890
// MI455X (gfx1250) — compile-verified
//
#include <hip/hip_runtime.h>

// Coulomb factor, rounded to f32 exactly as jnp does with the Python double.
#define KE_F 14.399645351950548f

typedef float v4f __attribute__((ext_vector_type(4)));
typedef int   v4i __attribute__((ext_vector_type(4)));

__device__ __forceinline__ float edge_energy(
    float r, float vx, float vy, float vz, int iu, int iv,
    const float* __restrict__ qml, const float* __restrict__ qmm,
    const float* __restrict__ dip, const float* __restrict__ quad)
{
    // Gathers: small tables, L2-resident (RT temporal hint).
    const float qu = qml[iu];
    const float qv = qmm[iv];
    const float* __restrict__ mu = dip + 3 * (size_t)iu;
    const float m0 = mu[0], m1 = mu[1], m2 = mu[2];
    const float* __restrict__ Q = quad + 9 * (size_t)iu;
    const float q00 = Q[0], q01 = Q[1], q02 = Q[2];
    const float q10 = Q[3], q11 = Q[4], q12 = Q[5];
    const float q20 = Q[6], q21 = Q[7], q22 = Q[8];

    const float rn  = 1.0f / r;       // B0 (IEEE div, matches XLA)
    const float rn2 = rn * rn;
    const float B1  = rn * rn2;       // 1/r^3
    const float B2  = 3.0f * B1 * rn2; // 3/r^5

    // dipole term: (mu . v) * q_v
    const float mudot = fmaf(m0, vx, fmaf(m1, vy, m2 * vz));

    // quadrupole term: sum_ij Q_ij (v_i v_j - delta_ij |v|^2/3)
    //                = v^T Q v - trQ * |v|^2 / 3
    const float vv3 = (vx * vx + vy * vy + vz * vz) * (1.0f / 3.0f);
    const float rx  = fmaf(q00, vx, fmaf(q01, vy, q02 * vz));
    const float ry  = fmaf(q10, vx, fmaf(q11, vy, q12 * vz));
    const float rz  = fmaf(q20, vx, fmaf(q21, vy, q22 * vz));
    const float vQv = fmaf(vx, rx, fmaf(vy, ry, vz * rz));
    const float trQ = q00 + q11 + q22;
    const float G2  = fmaf(-trQ, vv3, vQv);

    const float Eelec = rn * (qu * qv) + B1 * (mudot * qv) - B2 * (G2 * qv);
    return KE_F * Eelec;
}

__global__ __launch_bounds__(256) void mlmm_electrostatics_kernel(
    const float* __restrict__ dist,   // [E]
    const float* __restrict__ vec,    // [E,3]
    const float* __restrict__ qml,    // [N_ML]
    const float* __restrict__ qmm,    // [N_MM]
    const float* __restrict__ dip,    // [N_ML,3]
    const float* __restrict__ quad,   // [N_ML,9]
    const int*   __restrict__ idx_u,  // [E]
    const int*   __restrict__ idx_v,  // [E]
    float*       __restrict__ out,    // [E]
    int E)
{
    const int t  = blockIdx.x * blockDim.x + threadIdx.x;
    const int e0 = t * 4;
    if (e0 >= E) return;

    if (e0 + 4 <= E) {
        // Fully coalesced vector path: 4 consecutive edges per lane.
        // Non-temporal hints keep the 470MB streams from evicting the
        // ~1MB gather tables out of the 192MB L2.
        const v4f d  = __builtin_nontemporal_load((const v4f*)dist  + t);
        const v4i iu = __builtin_nontemporal_load((const v4i*)idx_u + t);
        const v4i iv = __builtin_nontemporal_load((const v4i*)idx_v + t);
        const v4f a  = __builtin_nontemporal_load((const v4f*)vec + 3 * (size_t)t + 0);
        const v4f b  = __builtin_nontemporal_load((const v4f*)vec + 3 * (size_t)t + 1);
        const v4f c  = __builtin_nontemporal_load((const v4f*)vec + 3 * (size_t)t + 2);

        // De-interleave [E,3] packed as 3 float4s -> per-edge (vx,vy,vz)
        const float vx[4] = { a.x, a.w, b.z, c.y };
        const float vy[4] = { a.y, b.x, b.w, c.z };
        const float vz[4] = { a.z, b.y, c.x, c.w };
        const float rr[4] = { d.x, d.y, d.z, d.w };
        const int   ius[4] = { iu.x, iu.y, iu.z, iu.w };
        const int   ivs[4] = { iv.x, iv.y, iv.z, iv.w };

        v4f o;
        #pragma unroll
        for (int k = 0; k < 4; ++k) {
            const float e = edge_energy(rr[k], vx[k], vy[k], vz[k],
                                        ius[k], ivs[k], qml, qmm, dip, quad);
            if (k == 0) o.x = e; else if (k == 1) o.y = e;
            else if (k == 2) o.z = e; else o.w = e;
        }
        __builtin_nontemporal_store(o, (v4f*)out + t);
    } else {
        // Tail (E not divisible by 4): scalar path.
        for (int e = e0; e < E; ++e) {
            const float r = dist[e];
            const float x = vec[3 * (size_t)e + 0];
            const float y = vec[3 * (size_t)e + 1];
            const float z = vec[3 * (size_t)e + 2];
            out[e] = edge_energy(r, x, y, z, idx_u[e], idx_v[e],
                                 qml, qmm, dip, quad);
        }
    }
}

extern "C" void kernel_launch(void* const* d_in, const int* in_sizes, int n_in,
                              void* d_out, int out_size, void* d_ws, size_t ws_size,
                              hipStream_t stream) {
    (void)n_in; (void)d_ws; (void)ws_size; (void)out_size;
    const float* dist = (const float*)d_in[0];   // mlmm_distances_uv [E]
    const float* vec  = (const float*)d_in[1];   // mlmm_vectors_uv   [E,3]
    const float* qml  = (const float*)d_in[2];   // atomic_charges    [N_ML]
    const float* qmm  = (const float*)d_in[3];   // mlmm_atomic_charges [N_MM]
    const float* dip  = (const float*)d_in[4];   // atomic_dipoles    [N_ML,3]
    const float* quad = (const float*)d_in[5];   // atomic_quadrupoles[N_ML,3,3]
    const int*   iu   = (const int*)d_in[6];     // mlmm_idx_u [E]
    const int*   iv   = (const int*)d_in[7];     // mlmm_idx_v [E]
    float*       out  = (float*)d_out;

    const int E = in_sizes[0];
    const int threads = 256;                 // 8 wave32 waves per block
    const int lanes   = (E + 3) / 4;         // 4 edges per lane
    const int blocks  = (lanes + threads - 1) / threads;

    mlmm_electrostatics_kernel<<<blocks, threads, 0, stream>>>(
        dist, vec, qml, qmm, dip, quad, iu, iv, out, E);
}